// BinaryEEGClassifier_90168543412331
// MI455X (gfx1250) — compile-verified
//
#include <hip/hip_runtime.h>
#include <hip/hip_bf16.h>

// ---------------- Izhikevich constants (from reference) ----------------
#define IZH_A      0.02f
#define IZH_TAUDT  0.25f          /* 250.0 * 0.001 */
#define IZH_TAUDTA 0.005f         /* TAU_DT * A    */
#define IZH_VTH    30.0f
#define IZH_C      (-65.0f)
#define IZH_D      6.0f
#define IZH_V0     (-70.0f)
#define IZH_U0     (-14.0f)

typedef __attribute__((ext_vector_type(2))) float v2f;
typedef __attribute__((ext_vector_type(8))) float v8f;

// One Izhikevich step; returns spike z in {0,1}, updates v,u in place.
__device__ __forceinline__ float izh_step(float i, float& v, float& u) {
    // v_ = v + TAU_DT*(0.04 v^2 + 5 v + 140 - u + i)
    float inner = fmaf(0.04f, v * v, fmaf(5.0f, v, (140.0f - u) + i));
    float v_ = fmaf(IZH_TAUDT, inner, v);
    // u_ = u + TAU_DT*A*(0.2 v - u)
    float u_ = fmaf(IZH_TAUDTA, fmaf(0.2f, v, -u), u);
    float z = (v_ - IZH_VTH > 0.0f) ? 1.0f : 0.0f;
    v = (z > 0.0f) ? IZH_C : v_;
    u = u_ + z * IZH_D;            // (1-z)u_ + z(u_+D) == u_ + zD
    return z;
}

// ---------------- kernel 0: fold W1,b1,W2,b2 into Weff[16][64], beff[2] ---
__global__ void snn_setup_weff(const float* __restrict__ W1, const float* __restrict__ b1,
                               const float* __restrict__ W2, const float* __restrict__ b2,
                               float* __restrict__ Weff, float* __restrict__ beff) {
    int tid = threadIdx.x;                 // 1024 threads, one per Weff entry
    int k = tid >> 6, c = tid & 63;
    float acc = 0.0f;
    if (k < 2) {
        #pragma unroll 8
        for (int j = 0; j < 64; ++j)
            acc = fmaf(W2[k * 64 + j], W1[j * 64 + c], acc);
    }
    Weff[tid] = acc;                       // rows k>=2 are zero padding
    if (tid < 2) {
        float bb = b2[tid];
        for (int j = 0; j < 64; ++j)
            bb = fmaf(W2[tid * 64 + j], b1[j], bb);
        beff[tid] = bb;
    }
}

// ---------------- kernel 1: H[k][m] = dot(Weff[k,:], X[m,:])  (WMMA) -----
// X: [Mtotal][64] f32.  One wave handles a 16-row tile; K=64 in 16 chunks of 4
// via V_WMMA_F32_16X16X4_F32 (A = Weff 16x4-chunk, B = X-tile^T 4x16-chunk).
#define P1_WAVES 8
#define LDS_STRIDE 68                     // 64 + 4 pad -> conflict-free frag reads

__global__ __launch_bounds__(256) void snn_proj_wmma(
        const float* __restrict__ X, const float* __restrict__ Weff,
        float* __restrict__ H, int Mtotal) {
    __shared__ float tile[P1_WAVES][16 * LDS_STRIDE + 4];

    const int lane = threadIdx.x & 31;
    const int wv   = threadIdx.x >> 5;
    const int tileIdx = blockIdx.x * P1_WAVES + wv;
    const int mbase = tileIdx * 16;
    if (mbase >= Mtotal) return;          // wave-uniform exit

    float* lds = &tile[wv][0];

    // Coalesced global->LDS: 16 rows x 64 floats (4 KB) via b128
    const float* src = X + (size_t)mbase * 64;
    #pragma unroll
    for (int it = 0; it < 8; ++it) {
        int flat = it * 128 + lane * 4;
        int r = flat >> 6, cl = flat & 63;
        float4 d = *(const float4*)(src + flat);
        *(float4*)(lds + r * LDS_STRIDE + cl) = d;
    }

    // Fragment lane mapping (ISA 32-bit A 16x4): M=lane&15, K = 2*(lane>>4)+{0,1}
    const int mrow = lane & 15;
    const int koff = (lane >> 4) * 2;

    // A fragments (Weff) for all 16 K-chunks: constant data, small gather
    v2f afrag[16];
    #pragma unroll
    for (int kc = 0; kc < 16; ++kc) {
        const float* ap = Weff + mrow * 64 + kc * 4 + koff;
        v2f a = {ap[0], ap[1]};
        afrag[kc] = a;
    }

    v8f acc = {};
    #pragma unroll
    for (int kc = 0; kc < 16; ++kc) {
        const float* bp = lds + mrow * LDS_STRIDE + kc * 4 + koff;  // n = mrow
        v2f bfrag = {bp[0], bp[1]};
#if defined(__HIP_DEVICE_COMPILE__)
        acc = __builtin_amdgcn_wmma_f32_16x16x4_f32(
                /*neg_a=*/false, afrag[kc], /*neg_b=*/false, bfrag,
                /*c_mod=*/(short)0, acc, /*reuse_a=*/false, /*reuse_b=*/false);
#else
        acc[0] = fmaf(afrag[kc][0], bfrag[0], acc[0]);  // host stub (never run)
#endif
    }

    // D layout: VGPR r, lanes 0-15 -> M=r, N=lane. Need M=0 (acc[0]) and M=1 (acc[1]).
    if (lane < 16) {
        H[mbase + lane]          = acc[0];
        H[Mtotal + mbase + lane] = acc[1];
    }
}

// ---------------- kernel 2: sequential Izhikevich chains ------------------
__global__ void snn_recurrence(const float* __restrict__ H, const float* __restrict__ beff,
                               const float* __restrict__ Wg2, const float* __restrict__ bg2,
                               const float* __restrict__ W3, const float* __restrict__ b3,
                               float* __restrict__ out, int T, int Bsz) {
    int b = blockIdx.x * blockDim.x + threadIdx.x;
    if (b >= Bsz) return;
    const float be0 = beff[0], be1 = beff[1];
    const float wg00 = Wg2[0], wg01 = Wg2[1], wg10 = Wg2[2], wg11 = Wg2[3];
    const float bgA = bg2[0], bgB = bg2[1];
    const float w30 = W3[0], w31 = W3[1], b3v = b3[0];

    float v10 = IZH_V0, u10 = IZH_U0, v11 = IZH_V0, u11 = IZH_U0;
    float v2  = IZH_V0, u2  = IZH_U0;

    const float* H0 = H;
    const float* H1 = H + (size_t)T * Bsz;

    for (int t = 0; t < T; ++t) {
        int m = t * Bsz + b;
        float i0 = H0[m] + be0;
        float i1 = H1[m] + be1;
        float z0 = izh_step(i0, v10, u10);
        float z1 = izh_step(i1, v11, u11);
        float g0 = fmaf(z0, wg00, fmaf(z1, wg01, bgA));
        float g1 = fmaf(z0, wg10, fmaf(z1, wg11, bgB));
        float g  = fmaf(g0, w30, fmaf(g1, w31, b3v));
        float z2 = izh_step(g, v2, u2);
        out[m] = z2;
    }
}

// ---------------- fallback: fused, no workspace needed --------------------
__global__ void snn_fused_fallback(const float* __restrict__ xs,
                                   const float* __restrict__ W1, const float* __restrict__ b1,
                                   const float* __restrict__ W2, const float* __restrict__ b2,
                                   const float* __restrict__ Wg2, const float* __restrict__ bg2,
                                   const float* __restrict__ W3, const float* __restrict__ b3,
                                   float* __restrict__ out, int T, int Bsz) {
    __shared__ float sW[2][64];
    __shared__ float sb[2];
    int tid = threadIdx.x;
    if (tid < 128) {
        int k = tid >> 6, c = tid & 63;
        float acc = 0.0f;
        for (int j = 0; j < 64; ++j) acc = fmaf(W2[k * 64 + j], W1[j * 64 + c], acc);
        sW[k][c] = acc;
    }
    if (tid < 2) {
        float bb = b2[tid];
        for (int j = 0; j < 64; ++j) bb = fmaf(W2[tid * 64 + j], b1[j], bb);
        sb[tid] = bb;
    }
    __syncthreads();

    int b = blockIdx.x * blockDim.x + tid;
    if (b >= Bsz) return;
    const float wg00 = Wg2[0], wg01 = Wg2[1], wg10 = Wg2[2], wg11 = Wg2[3];
    const float bgA = bg2[0], bgB = bg2[1];
    const float w30 = W3[0], w31 = W3[1], b3v = b3[0];
    float v10 = IZH_V0, u10 = IZH_U0, v11 = IZH_V0, u11 = IZH_U0;
    float v2  = IZH_V0, u2  = IZH_U0;

    for (int t = 0; t < T; ++t) {
        const float* xr = xs + ((size_t)t * Bsz + b) * 64;
        float h0 = sb[0], h1 = sb[1];
        #pragma unroll 8
        for (int c = 0; c < 64; ++c) {
            float x = xr[c];
            h0 = fmaf(x, sW[0][c], h0);
            h1 = fmaf(x, sW[1][c], h1);
        }
        float z0 = izh_step(h0, v10, u10);
        float z1 = izh_step(h1, v11, u11);
        float g0 = fmaf(z0, wg00, fmaf(z1, wg01, bgA));
        float g1 = fmaf(z0, wg10, fmaf(z1, wg11, bgB));
        float g  = fmaf(g0, w30, fmaf(g1, w31, b3v));
        out[t * Bsz + b] = izh_step(g, v2, u2);
    }
}

// ---------------- host entry ---------------------------------------------
extern "C" void kernel_launch(void* const* d_in, const int* in_sizes, int n_in,
                              void* d_out, int out_size, void* d_ws, size_t ws_size,
                              hipStream_t stream) {
    (void)n_in;
    const float* xs  = (const float*)d_in[0];
    const float* W1  = (const float*)d_in[1];
    const float* b1  = (const float*)d_in[2];
    const float* W2  = (const float*)d_in[3];
    const float* b2  = (const float*)d_in[4];
    const float* Wg2 = (const float*)d_in[5];
    const float* bg2 = (const float*)d_in[6];
    const float* W3  = (const float*)d_in[7];
    const float* b3  = (const float*)d_in[8];
    float* out = (float*)d_out;

    const int C = 64;
    const int Mtotal = in_sizes[0] / C;     // T * B  (== out_size)
    const int Bsz = 512;
    const int T = Mtotal / Bsz;

    // ws layout: Weff[16*64] | beff[16 (2 used)] | H[2*Mtotal]
    const size_t need = (size_t)(16 * 64 + 16 + 2 * (size_t)Mtotal) * sizeof(float);

    if (ws_size >= need && (Mtotal % 16) == 0 && out_size == Mtotal) {
        float* Weff = (float*)d_ws;
        float* beff = Weff + 16 * 64;
        float* H    = beff + 16;

        snn_setup_weff<<<1, 1024, 0, stream>>>(W1, b1, W2, b2, Weff, beff);

        const int tiles  = Mtotal / 16;
        const int blocks = (tiles + P1_WAVES - 1) / P1_WAVES;
        snn_proj_wmma<<<blocks, 256, 0, stream>>>(xs, Weff, H, Mtotal);

        snn_recurrence<<<(Bsz + 255) / 256, 256, 0, stream>>>(
            H, beff, Wg2, bg2, W3, b3, out, T, Bsz);
    } else {
        snn_fused_fallback<<<(Bsz + 255) / 256, 256, 0, stream>>>(
            xs, W1, b1, W2, b2, Wg2, bg2, W3, b3, out, T, Bsz);
    }
}